// PerturbedTopK_24988119728670
// MI455X (gfx1250) — compile-verified
//
#include <hip/hip_runtime.h>
#include <cstdint>

// Perturbed top-k for MI455X (gfx1250, wave32).
// One wave per (b,n) row: async global->LDS stream of the noise row,
// exact 4-pass radix select of the k-th largest key (LDS histograms),
// candidate compaction + O(c^2/32) rank-by-count, atomic scatter of 1/n.

typedef unsigned int       u32;
typedef unsigned long long u64;

#define B_      16
#define N_      1000
#define T_      2048
#define K_      204
#define SIGMA_  0.05f
#define INVN_   (1.0f / 1000.0f)
#define WAVES_  8
#define TPB_    256
#define CAP_    256          // candidate capacity (>= K_; ties beyond are vanishingly rare)
#define JMAX_   (CAP_ / 32)  // candidate slots per lane
#define WBYTES_ 11264        // per-wave LDS: 8192 keys + 1024 hist + 2048 cand

#define CBAR() asm volatile("" ::: "memory")

// The gfx1250 async-load builtin takes v4i pointers in AS1 (global) / AS3 (LDS).
typedef int v4i __attribute__((ext_vector_type(4)));
typedef __attribute__((address_space(1))) v4i as1_v4i;
typedef __attribute__((address_space(3))) v4i as3_v4i;

__device__ __forceinline__ void async_ldg_b128(const void* g, void* l) {
#if __has_builtin(__builtin_amdgcn_global_load_async_to_lds_b128)
    __builtin_amdgcn_global_load_async_to_lds_b128(
        (as1_v4i*)(uintptr_t)g,
        (as3_v4i*)(u32)(uintptr_t)l,   // low 32 bits of a generic LDS pointer == LDS offset
        0, 0);
#else
    *(uint4*)l = *(const uint4*)g;     // fallback: global_load_b128 + ds_store_b128
#endif
}

__device__ __forceinline__ void async_wait0() {
#if __has_builtin(__builtin_amdgcn_s_wait_asynccnt)
    __builtin_amdgcn_s_wait_asynccnt(0);
#else
    asm volatile("s_wait_asynccnt 0x0" ::: "memory");
#endif
}

__global__ __launch_bounds__(TPB_)
void ptk_kernel(const float* __restrict__ x,
                const float* __restrict__ noise,
                float* __restrict__ out) {
    extern __shared__ __align__(16) unsigned char smem[];
    const u32 lane = threadIdx.x & 31u;
    const u32 wave = threadIdx.x >> 5;
    const u32 row  = blockIdx.x * WAVES_ + wave;   // row = b*N_ + n
    if (row >= (u32)(B_ * N_)) return;             // uniform per wave
    const u32 b = row / N_;

    unsigned char* wb = smem + (size_t)wave * WBYTES_;
    u32* keysb = (u32*)wb;                 // 2048 x u32 (noise bits, then sort keys)
    u32* hist  = (u32*)(wb + 8192);        // 256 bins
    u64* cand  = (u64*)(wb + 9216);        // CAP_ packed sortkeys

    // ---- Phase 1: async copy one 8KB noise row into LDS (ASYNCcnt path) ----
    const char* gsrc = (const char*)noise + (size_t)row * (T_ * 4);
    char* lbase = (char*)keysb;
    #pragma unroll
    for (int i = 0; i < 16; ++i) {
        int off = i * 512 + (int)lane * 16;        // 32 lanes x 16B = 512B / issue
        async_ldg_b128(gsrc + off, lbase + off);
    }
    async_wait0();
    CBAR();

    // ---- Phase 2: in-place transform to monotonic u32 sort keys ----
    const float* xrow = x + (size_t)b * T_;
    #pragma unroll 8
    for (int i = 0; i < 64; ++i) {
        u32 d = (u32)i * 32u + lane;
        float f = fmaf(__uint_as_float(keysb[d]), SIGMA_, xrow[d]);
        u32 u = __float_as_uint(f);
        keysb[d] = (u >> 31) ? ~u : (u | 0x80000000u);  // unsigned order == float order
    }
    CBAR();

    // ---- Phase 3: exact radix select of the K-th largest key ----
    u32 kk = K_, prefix = 0u;
    #pragma unroll
    for (int pass = 0; pass < 4; ++pass) {
        const int shift = 24 - 8 * pass;
        const u32 pmask = (pass == 0) ? 0u : (0xFFFFFFFFu << (shift + 8));

        for (u32 j = lane; j < 256u; j += 32u) hist[j] = 0u;
        CBAR();
        for (int i = 0; i < 64; ++i) {
            u32 kv = keysb[(u32)i * 32u + lane];
            if ((kv & pmask) == prefix)
                atomicAdd(&hist[(kv >> shift) & 255u], 1u);   // ds_add_u32
        }
        CBAR();
        // suffix scan from the top bins; lane l owns bins [255-8l .. 248-8l]
        u32 binTop = 255u - 8u * lane;
        u32 cnts[8]; u32 s = 0u;
        #pragma unroll
        for (int j = 0; j < 8; ++j) { cnts[j] = hist[binTop - (u32)j]; s += cnts[j]; }
        u32 incl = s;
        #pragma unroll
        for (int off = 1; off < 32; off <<= 1) {
            u32 up = __shfl_up(incl, (unsigned)off);
            if ((int)lane >= off) incl += up;
        }
        u32 excl = incl - s;
        bool sel = (excl < kk) && (kk <= incl);     // exactly one lane
        u32 selmask = (u32)__ballot(sel);
        int srclane = __ffs((int)selmask) - 1;
        u32 digit = 0u, nk = kk;
        if (sel) {
            u32 run = excl;
            #pragma unroll
            for (int j = 0; j < 8; ++j) {
                if (run + cnts[j] >= kk) { digit = binTop - (u32)j; nk = kk - run; break; }
                run += cnts[j];
            }
        }
        digit = (u32)__shfl((int)digit, srclane);
        kk    = (u32)__shfl((int)nk, srclane);
        prefix |= digit << shift;
        CBAR();
    }
    const u32 Tkey = prefix;   // exact K-th largest key value

    // ---- Phase 4: compact candidates (key >= Tkey) into LDS ----
    // sortkey = (key << 11) | (2047 - idx): strict u64 '>' reproduces
    // descending-value order with lower-index-wins tie-break (jax.lax.top_k).
    u32 cbase = 0u;
    for (int i = 0; i < 64; ++i) {
        u32 d  = (u32)i * 32u + lane;
        u32 kv = keysb[d];
        bool pred = (kv >= Tkey);
        u32 m = (u32)__ballot(pred);
        if (pred) {
            u32 pos = cbase + (u32)__popc(m & ((1u << lane) - 1u));
            if (pos < (u32)CAP_)
                cand[pos] = ((u64)kv << 11) | (u64)(2047u - d);
        }
        cbase += (u32)__popc(m);
    }
    u32 c = cbase < (u32)CAP_ ? cbase : (u32)CAP_;
    CBAR();

    // ---- Phase 5: rank by counting among candidates, scatter 1/n ----
    u64 me[JMAX_]; u32 rk[JMAX_];
    #pragma unroll
    for (int jj = 0; jj < JMAX_; ++jj) {
        u32 j = lane + 32u * (u32)jj;
        me[jj] = (j < c) ? cand[j] : 0ull;
        rk[jj] = 0u;
    }
    for (u32 m2 = 0; m2 < c; ++m2) {
        u64 cm = cand[m2];                       // uniform-address LDS broadcast
        #pragma unroll
        for (int jj = 0; jj < JMAX_; ++jj) rk[jj] += (cm > me[jj]) ? 1u : 0u;
    }
    #pragma unroll
    for (int jj = 0; jj < JMAX_; ++jj) {
        u32 j = lane + 32u * (u32)jj;
        if (j < c && rk[jj] < (u32)K_) {
            u32 d = 2047u - (u32)(me[jj] & 0x7FFull);
            atomicAdd(out + ((size_t)b * K_ + rk[jj]) * T_ + d, INVN_);
        }
    }
}

__global__ void zero_kernel(float* __restrict__ out, int n) {
    int i = (blockIdx.x * blockDim.x + threadIdx.x) * 4;
    if (i + 3 < n) {
        *(float4*)(out + i) = make_float4(0.f, 0.f, 0.f, 0.f);
    } else {
        for (int j = i; j < n; ++j) out[j] = 0.f;
    }
}

extern "C" void kernel_launch(void* const* d_in, const int* in_sizes, int n_in,
                              void* d_out, int out_size, void* d_ws, size_t ws_size,
                              hipStream_t stream) {
    const float* x     = (const float*)d_in[0];
    const float* noise = (const float*)d_in[1];
    float* out = (float*)d_out;

    // Zero the indicator buffer (harness poisons it; atomics accumulate into it).
    int zblocks = (out_size + TPB_ * 4 - 1) / (TPB_ * 4);
    zero_kernel<<<zblocks, TPB_, 0, stream>>>(out, out_size);

    // 16000 rows, 8 waves (rows) per 256-thread workgroup -> 2000 WGs.
    dim3 grid((B_ * N_) / WAVES_);
    dim3 block(TPB_);
    size_t lds_bytes = (size_t)WAVES_ * WBYTES_;   // 88 KB dynamic LDS
    ptk_kernel<<<grid, block, lds_bytes, stream>>>(x, noise, out);
}